// FastfoodTransform_44255343018472
// MI455X (gfx1250) — compile-verified
//
#include <hip/hip_runtime.h>

// Fastfood transform, MI455X (gfx1250, wave32).
// LL = 2^23, d = 1024.  The first big FWHT collapses to FWHT_1024 (input
// support is only the low 10 index bits):  y = FWHT1024(BB*x);
// t[i] = GG[i]*y[Pi[i]&1023]; out = FWHT_{2^23}(t) done as
// (13 low-bit stages fused with the gather) + (10 high-bit stages in-place).
// All radix-16 sub-transforms run on the matrix pipe via
// V_WMMA_F32_16X16X4_F32 (4 chained K=4 WMMAs = one 16x16 Hadamard matmul
// = 4 butterfly stages, exact f32 math).  Memory-bound: >= 96 MB HBM
// traffic (~4.1 us @ 23.3 TB/s); the 32 MB intermediate stays in L2.

#define LL_LOG2 23
#define LL (1u << LL_LOG2)

typedef float f32x2 __attribute__((ext_vector_type(2)));
typedef float f32x8 __attribute__((ext_vector_type(8)));

__device__ __forceinline__ float hval(int p, int q) {
    // Walsh-Hadamard entry H[p][q] = (-1)^popcount(p & q)
    return (__builtin_popcount(p & q) & 1) ? -1.0f : 1.0f;
}

// Precompute the A-operand fragments of H16 (depends only on lane id).
// A-layout (32-bit A 16x4): lane l, reg r -> M = l&15, K = r + 2*(l>>4).
__device__ __forceinline__ void make_h16_afrags(int lane, f32x2 aH[4]) {
    const int m16 = lane & 15;
    const int kofs = 2 * (lane >> 4);
    for (int j = 0; j < 4; ++j) {
        aH[j].x = hval(m16, 4 * j + kofs);
        aH[j].y = hval(m16, 4 * j + kofs + 1);
    }
}

// Apply X <- H16 * X along the "row" dimension of a 16x16 matrix embedded
// in LDS at  addr(row,col) = base[row*rs + col*ns]  (strides in floats).
// B-layout (4x16): lane l, reg r -> K = r + 2*(l>>4), N = l&15.
// C/D-layout (16x16): lane l, reg r -> M = r + 8*(l>>4), N = l&15.
__device__ __forceinline__ void wmma_h16_pass(float* base, int rs, int ns,
                                              int lane, const f32x2 aH[4]) {
    const int n = lane & 15;
    const int kofs = 2 * (lane >> 4);
    f32x8 acc = {};
    for (int j = 0; j < 4; ++j) {
        f32x2 b;
        const int k = 4 * j + kofs;
        b.x = base[k * rs + n * ns];
        b.y = base[(k + 1) * rs + n * ns];
        acc = __builtin_amdgcn_wmma_f32_16x16x4_f32(
            false, aH[j], false, b, (short)0, acc, false, false);
    }
    const int m0 = 8 * (lane >> 4);
    for (int r = 0; r < 8; ++r)
        base[(r + m0) * rs + n * ns] = acc[r];
}

// ---------------------------------------------------------------------------
// K1: y = FWHT_1024(BB[0:1024] * x)   (single block, 256 threads; tiny)
// ---------------------------------------------------------------------------
__global__ void __launch_bounds__(256)
k1_fwht1024(const float* __restrict__ x, const float* __restrict__ BB,
            float* __restrict__ y) {
    __shared__ float s[1024];
    const int t = threadIdx.x;
    for (int k = 0; k < 4; ++k) {
        int j = t + 256 * k;
        s[j] = BB[j] * x[j];
    }
    __syncthreads();
    for (int h = 1; h < 1024; h <<= 1) {
        for (int p = t; p < 512; p += 256) {
            int i = ((p & ~(h - 1)) << 1) | (p & (h - 1));
            float a = s[i], b = s[i + h];
            s[i] = a + b;
            s[i + h] = a - b;
        }
        __syncthreads();
    }
    for (int k = 0; k < 4; ++k) {
        int j = t + 256 * k;
        y[j] = s[j];
    }
}

// ---------------------------------------------------------------------------
// K2: gather/scale + FWHT over the low 13 bits (8192-element blocks).
//     Index in block: i = g*256 + b*16 + c   (g: 5 bits, b,c: 4 bits).
//     bits 0-3 (c), 4-7 (b), 8-11 (g&15) via WMMA; bit 12 scalar.
//     1024 blocks x 256 threads (8 wave32).
// ---------------------------------------------------------------------------
__global__ void __launch_bounds__(256)
k2_gather_fwht8192(const float* __restrict__ y, const int* __restrict__ Pi,
                   const float* __restrict__ GG, float* __restrict__ out) {
    __shared__ float ytile[1024];
    __shared__ float data[8192];
    const int t = threadIdx.x;
    const unsigned base = blockIdx.x * 8192u;
    const int lane = t & 31;
    const int wv = t >> 5;  // wave id 0..7

    f32x2 aH[4];
    make_h16_afrags(lane, aH);

    // stage y into LDS (it is read 8192x per block)
    for (int k = 0; k < 4; ++k) ytile[t + 256 * k] = y[t + 256 * k];
    __syncthreads();

    // gather: t[i] = GG[i] * y[Pi[i] & 1023]  (coalesced b32 streams)
    for (int k = 0; k < 32; ++k) {
        unsigned i = base + (unsigned)t + 256u * k;
        data[t + 256 * k] = GG[i] * ytile[Pi[i] & 1023];
    }
    __syncthreads();

    // bits 0-3 (rows c, cols b) and bits 4-7 (rows b, cols c):
    // both stay inside one wave-private g-slice -> no barrier between them.
    for (int s = 0; s < 4; ++s) {
        float* X = &data[(wv * 4 + s) * 256];
        wmma_h16_pass(X, /*rs=*/1, /*ns=*/16, lane, aH);   // FWHT on c
        wmma_h16_pass(X, /*rs=*/16, /*ns=*/1, lane, aH);   // FWHT on b
    }
    __syncthreads();

    // bits 8-11: rows gl = g&15 (stride 256), cols c; matrix per (g>>4, b)
    for (int s = 0; s < 4; ++s) {
        const int idx = wv * 4 + s;          // 0..31
        const int g4 = idx >> 4;             // g bit 4
        const int b = idx & 15;
        wmma_h16_pass(&data[g4 * 4096 + b * 16], /*rs=*/256, /*ns=*/1,
                      lane, aH);
    }
    __syncthreads();

    // bit 12: single scalar butterfly stage, stride 4096
    for (int p = t; p < 4096; p += 256) {
        float a = data[p], b = data[p + 4096];
        data[p] = a + b;
        data[p + 4096] = a - b;
    }
    __syncthreads();

    for (int k = 0; k < 32; ++k)
        out[base + (unsigned)t + 256u * k] = data[t + 256 * k];
}

// ---------------------------------------------------------------------------
// K3: FWHT over the high 10 bits (global strides 8192..2^22), in-place.
//     Each block owns 16 consecutive low-offsets l and all 1024 strided
//     elements per l (disjoint across blocks -> in-place is safe).
//     LDS tile m[h*16 + l], h: 10 bits.  h bits 0-3 and 4-7 via WMMA,
//     h bits 8-9 scalar.  512 blocks x 256 threads, 64 KB LDS.
// ---------------------------------------------------------------------------
__global__ void __launch_bounds__(256)
k3_fwht_high(float* __restrict__ out) {
    __shared__ float m[16384];            // 1024 h x 16 l
    const int t = threadIdx.x;
    const unsigned l0 = blockIdx.x * 16u;
    const int lane = t & 31;
    const int wv = t >> 5;
    const int l = t & 15;
    const int hb = t >> 4;                // 0..15

    f32x2 aH[4];
    make_h16_afrags(lane, aH);

    for (int hh = hb; hh < 1024; hh += 16)
        m[hh * 16 + l] = out[(unsigned)hh * 8192u + l0 + l];
    __syncthreads();

    // h bits 0-3: rows h&15 (stride 16), cols l; matrix per h>>4 (64 total)
    for (int s = 0; s < 8; ++s)
        wmma_h16_pass(&m[(wv * 8 + s) * 256], /*rs=*/16, /*ns=*/1, lane, aH);
    __syncthreads();

    // h bits 4-7: rows (h>>4)&15 (stride 256), cols l;
    // matrix per (h>>8, h&15) (64 total)
    for (int s = 0; s < 8; ++s) {
        const int idx = wv * 8 + s;        // 0..63
        const int h8 = idx >> 4;           // h bits 8-9
        const int hl = idx & 15;           // h bits 0-3
        wmma_h16_pass(&m[h8 * 4096 + hl * 16], /*rs=*/256, /*ns=*/1,
                      lane, aH);
    }
    __syncthreads();

    // h bits 8-9: two scalar stages, LDS strides 4096 and 8192
    for (int st = 4096; st <= 8192; st <<= 1) {
        for (int q = t; q < 8192; q += 256) {
            int i = ((q & ~(st - 1)) << 1) | (q & (st - 1));
            float a = m[i], b = m[i + st];
            m[i] = a + b;
            m[i + st] = a - b;
        }
        __syncthreads();
    }

    for (int hh = hb; hh < 1024; hh += 16)
        out[(unsigned)hh * 8192u + l0 + l] = m[hh * 16 + l];
}

// ---------------------------------------------------------------------------
extern "C" void kernel_launch(void* const* d_in, const int* in_sizes, int n_in,
                              void* d_out, int out_size, void* d_ws,
                              size_t ws_size, hipStream_t stream) {
    const float* x  = (const float*)d_in[0];   // (1024,) f32
    const float* BB = (const float*)d_in[1];   // (LL,)   f32, +/-1
    const float* GG = (const float*)d_in[2];   // (LL,)   f32
    const int*   Pi = (const int*)d_in[3];     // (LL,)   i32 permutation
    float* out = (float*)d_out;                // (LL,)   f32
    float* y = (float*)d_ws;                   // 1024 floats of scratch

    k1_fwht1024<<<1, 256, 0, stream>>>(x, BB, y);
    k2_gather_fwht8192<<<LL / 8192, 256, 0, stream>>>(y, Pi, GG, out);
    k3_fwht_high<<<8192 / 16, 256, 0, stream>>>(out);
}